// ManualGCNLayer_39908836115041
// MI455X (gfx1250) — compile-verified
//
#include <hip/hip_runtime.h>
#include <math.h>

#define CH 128  // feature channels, in == out == 128

typedef __attribute__((ext_vector_type(2))) float v2f;
typedef __attribute__((ext_vector_type(8))) float v8f;

// ---------------------------------------------------------------- zero scratch
__global__ void gcn_zero_kernel(float* __restrict__ p, int n) {
  int i = (int)blockIdx.x * (int)blockDim.x + (int)threadIdx.x;
  if (i < n) p[i] = 0.0f;
}

// ---------------------------------------------------------------- degree count
__global__ void gcn_degree_kernel(const int* __restrict__ row,
                                  float* __restrict__ deg, int E) {
  int e = (int)blockIdx.x * (int)blockDim.x + (int)threadIdx.x;
  if (e < E) atomicAdd(&deg[row[e]], 1.0f);
}

// ---------------------------------------------------------------- deg^{-1/2}
__global__ void gcn_dis_kernel(const float* __restrict__ deg,
                               float* __restrict__ dis, int n) {
  int i = (int)blockIdx.x * (int)blockDim.x + (int)threadIdx.x;
  if (i < n) {
    float d = deg[i];
    dis[i] = (d > 0.0f) ? (1.0f / sqrtf(d)) : 0.0f;
  }
}

// ---------------------------------------------------------------- edge scatter
// One wave32 per edge: lane l handles channels [4l, 4l+3].
// agg[row] += x[col] * dis[row]*dis[col]
__global__ void gcn_scatter_kernel(const float* __restrict__ x,
                                   const int* __restrict__ ei,
                                   const float* __restrict__ dis,
                                   float* __restrict__ agg, int E) {
  int e = (int)blockIdx.x * 8 + ((int)threadIdx.x >> 5);
  if (e >= E) return;
  int lane = (int)threadIdx.x & 31;
  int r = ei[e];       // destination (segment id)
  int c = ei[E + e];   // source node
  float norm = dis[r] * dis[c];
  float4 v = ((const float4*)(x + (size_t)c * CH))[lane];
  float* dst = agg + (size_t)r * CH + lane * 4;
  atomicAdd(dst + 0, v.x * norm);
  atomicAdd(dst + 1, v.y * norm);
  atomicAdd(dst + 2, v.z * norm);
  atomicAdd(dst + 3, v.w * norm);
}

// ---------------------------------------------------------------- WMMA GEMM
// out = agg @ W^T + b, full f32 via V_WMMA_F32_16X16X4_F32.
// One wave computes one 16x16 output tile; 8 waves/block cover all 8 n-tiles.
__global__ void gcn_gemm_wmma_kernel(const float* __restrict__ agg,
                                     const float* __restrict__ W,
                                     const float* __restrict__ bias,
                                     float* __restrict__ out, int nRows) {
  const int lane   = (int)threadIdx.x & 31;
  const int wave   = (int)threadIdx.x >> 5;   // n-tile index, 0..7
  const int m0     = (int)blockIdx.x * 16;
  const int n0     = wave * 16;
  const int laneHi = lane >> 4;               // 0: lanes 0-15, 1: lanes 16-31
  const int l15    = lane & 15;
  const int n      = n0 + l15;                // output column this lane owns

  // C/D layout: VGPR v <-> row m0+v (lanes 0-15) / m0+v+8 (lanes 16-31),
  // column = n0 + (lane&15). Bias depends only on the column -> seed C with it.
  v8f c;
  float bv = bias[n];
#pragma unroll
  for (int v = 0; v < 8; ++v) c[v] = bv;

  int rowA = m0 + l15;                        // A row this lane streams
  if (rowA >= nRows) rowA = nRows - 1;        // clamp (nRows % 16 == 0 here)
  const float* aptr = agg + (size_t)rowA * CH;
  const float* bptr = W + (size_t)n * CH;     // B[k][n] = W[n][k]

  // f32 A 16x4 layout: lanes 0-15 hold K=k,k+1; lanes 16-31 hold K=k+2,k+3.
  // B 4x16 is the transposed analog per-lane.
#pragma unroll
  for (int k = 0; k < CH; k += 4) {
    const int kk = k + laneHi * 2;
    v2f a, bf;
    a.x  = aptr[kk];
    a.y  = aptr[kk + 1];
    bf.x = bptr[kk];
    bf.y = bptr[kk + 1];
    c = __builtin_amdgcn_wmma_f32_16x16x4_f32(
        /*neg_a=*/false, a, /*neg_b=*/false, bf,
        /*c_mod=*/(short)0, c, /*reuse_a=*/false, /*reuse_b=*/false);
  }

#pragma unroll
  for (int v = 0; v < 8; ++v) {
    int r = m0 + v + laneHi * 8;
    if (r < nRows) out[(size_t)r * CH + n] = c[v];
  }
}

// ---------------------------------------------------------------- launch
extern "C" void kernel_launch(void* const* d_in, const int* in_sizes, int n_in,
                              void* d_out, int out_size, void* d_ws, size_t ws_size,
                              hipStream_t stream) {
  const float* x  = (const float*)d_in[0];   // [N, 128]
  const int*   ei = (const int*)d_in[1];     // [2, E]
  const float* W  = (const float*)d_in[2];   // [128, 128]
  const float* b  = (const float*)d_in[3];   // [128]
  float* out = (float*)d_out;                // [N, 128]

  const int N = in_sizes[0] / CH;
  const int E = in_sizes[1] / 2;

  // Scratch layout (floats): agg[N*CH] | deg[N] | dis[N]  (~26 MB)
  float* agg = (float*)d_ws;
  float* deg = agg + (size_t)N * CH;
  float* dis = deg + N;

  const int zn = N * CH + N;  // zero agg + deg contiguously
  gcn_zero_kernel<<<(zn + 255) / 256, 256, 0, stream>>>(agg, zn);
  gcn_degree_kernel<<<(E + 255) / 256, 256, 0, stream>>>(ei, deg, E);
  gcn_dis_kernel<<<(N + 255) / 256, 256, 0, stream>>>(deg, dis, N);
  gcn_scatter_kernel<<<(E + 7) / 8, 256, 0, stream>>>(x, ei, dis, agg, E);

  const int mTiles = (N + 15) / 16;  // 3125 for N=50000
  gcn_gemm_wmma_kernel<<<mTiles, 256, 0, stream>>>(agg, W, b, out, N);
}